// AutoregressiveLSTM_61881888800918
// MI455X (gfx1250) — compile-verified
//
#include <hip/hip_runtime.h>

typedef float v2f __attribute__((ext_vector_type(2)));
typedef float v8f __attribute__((ext_vector_type(8)));

#define BATCH 16384
#define TT    100
#define INF   4
#define HIDN  10
#define OUTF  4
#define PP    5
#define TPN   96   // TT - PP + 1

// ---------- fast tanh (native v_tanh_f32 on gfx1250; verified) --------------
__device__ __forceinline__ float fast_tanh(float x) {
#if __has_builtin(__builtin_amdgcn_tanhf)
    return __builtin_amdgcn_tanhf(x);
#else
    float e = __builtin_amdgcn_exp2f(x * -2.885390082f); // exp(-2x)
    return __builtin_amdgcn_rcpf(1.0f + e) * 2.0f - 1.0f;
#endif
}

// ---------- one LSTM cell: gates via 16 x V_WMMA_F32_16X16X4_F32 -----------
// Pre-scaling / padding contract:
//   * i/f/o rows of W_ih/W_hh and biases pre-scaled by 0.5 -> sigmoid = 0.5*tanh(acc)+0.5
//   * h-columns of W_hh and W_lin pre-scaled by 0.5        -> hD holds 2*h
//   * K padding feature 14 carries constant 1.0 in A and the (pre-scaled)
//     bias in B, so bias is added by the matmul; C starts as inline 0.
__device__ __forceinline__ void lstm_cell(const v2f a[4], const v2f bw[4][4],
                                          v8f& c, v8f& hD) {
    const v8f z = {};
    v8f acc0 = __builtin_amdgcn_wmma_f32_16x16x4_f32(false, a[0], false, bw[0][0], (short)0, z, false, false);
    v8f acc1 = __builtin_amdgcn_wmma_f32_16x16x4_f32(false, a[0], false, bw[1][0], (short)0, z, false, false);
    v8f acc2 = __builtin_amdgcn_wmma_f32_16x16x4_f32(false, a[0], false, bw[2][0], (short)0, z, false, false);
    v8f acc3 = __builtin_amdgcn_wmma_f32_16x16x4_f32(false, a[0], false, bw[3][0], (short)0, z, false, false);
#pragma unroll
    for (int s = 1; s < 4; ++s) {
        acc0 = __builtin_amdgcn_wmma_f32_16x16x4_f32(false, a[s], false, bw[0][s], (short)0, acc0, false, false);
        acc1 = __builtin_amdgcn_wmma_f32_16x16x4_f32(false, a[s], false, bw[1][s], (short)0, acc1, false, false);
        acc2 = __builtin_amdgcn_wmma_f32_16x16x4_f32(false, a[s], false, bw[2][s], (short)0, acc2, false, false);
        acc3 = __builtin_amdgcn_wmma_f32_16x16x4_f32(false, a[s], false, bw[3][s], (short)0, acc3, false, false);
    }
#pragma unroll
    for (int k = 0; k < 8; ++k) {
        float ui = fast_tanh(acc0[k]);          // acc_i pre-scaled by 0.5
        float uf = fast_tanh(acc1[k]);          // acc_f pre-scaled by 0.5
        float tg = fast_tanh(acc2[k]);          // g gate: full scale
        float uo = fast_tanh(acc3[k]);          // acc_o pre-scaled by 0.5
        float si = fmaf(0.5f, ui, 0.5f);
        float sf = fmaf(0.5f, uf, 0.5f);
        float cn = fmaf(sf, c[k], si * tg);     // true-valued c
        c[k] = cn;
        float tc = fast_tanh(cn);
        hD[k] = fmaf(uo, tc, tc);               // = (1+tanh(acc_o))*tanh(c) = 2*h
    }
}

// ---------- D-layout h -> A-layout a[1..3] via per-wave LDS transpose -------
// a[3] half=1 lanes carry the bias-activation constant {1,0} (K padding).
__device__ __forceinline__ void transpose_h(const v8f& hD, float* buf,
                                            int j, int half, v2f a[4]) {
#pragma unroll
    for (int k = 0; k < 8; ++k)
        buf[(k + 8 * half) * 18 + j] = hD[k];          // buf[m][hidden]
    asm volatile("s_wait_dscnt 0x0" ::: "memory");     // single-wave, in-order DS
    a[1] = *(const v2f*)(buf + j * 18 + 2 * half);     // h(2h), h(2h+1)
    a[2] = *(const v2f*)(buf + j * 18 + 4 + 2 * half); // h(4+2h), h(5+2h)
    v2f t3 = *(const v2f*)(buf + j * 18 + 8);          // h8, h9 (half=0 slots)
    a[3] = half ? v2f{1.0f, 0.0f} : t3;                // half=1: bias lane
}

// ---------- y = h @ W_lin^T + b_lin (wl pre-scaled 0.5 for 2x-h) -----------
__device__ __forceinline__ void lin_out(const v2f a[4], const float wl[4][3][2],
                                        const float blin[4], float y[4]) {
#pragma unroll
    for (int n = 0; n < 4; ++n) {
        float p = 0.0f;
#pragma unroll
        for (int s = 1; s < 4; ++s) {
            p = fmaf(a[s].x, wl[n][s - 1][0], p);
            p = fmaf(a[s].y, wl[n][s - 1][1], p);
        }
        int pi = __builtin_amdgcn_ds_swizzle(__builtin_bit_cast(int, p), 0x401F); // SWAPX16
        y[n] = blin[n] + p + __builtin_bit_cast(float, pi);
    }
}

__global__ __launch_bounds__(128, 1)
void arlstm_kernel(const float* __restrict__ x,
                   const float* __restrict__ W_ih, const float* __restrict__ W_hh,
                   const float* __restrict__ b_ih, const float* __restrict__ b_hh,
                   const float* __restrict__ W_lin, const float* __restrict__ b_lin,
                   float* __restrict__ out_y, float* __restrict__ out_c) {
    __shared__ __align__(16) float hbuf[4][16 * 18];

    const int lane = threadIdx.x & 31;
    const int wv   = threadIdx.x >> 5;
    const int j    = lane & 15;        // gate/hidden index in D layout; batch row in A layout
    const int half = lane >> 4;
    const long m0  = (long)(blockIdx.x * 4 + wv) * 16;

    // ---- build B matrices (4x16, K x N) per gate group / K-step, in regs ----
    // B layout mirrors A: lane -> n=lane%16, v2f = {K(2h), K(2h+1)}
    v2f bw[4][4];
#pragma unroll
    for (int G = 0; G < 4; ++G) {
        const float rs = (G == 2) ? 1.0f : 0.5f;   // sigmoid row pre-scale (i,f,o)
        const float hs = rs * 0.5f;                // + h-column pre-scale (stored h is 2x)
        const bool valid = (j < HIDN);
        const int  row   = G * HIDN + (valid ? j : 0);
        float w0 = valid ? rs * W_ih[row * INF + 2 * half]     : 0.0f;
        float w1 = valid ? rs * W_ih[row * INF + 2 * half + 1] : 0.0f;
        bw[G][0] = v2f{w0, w1};
#pragma unroll
        for (int s = 1; s < 3; ++s) {
            int k0 = 4 * (s - 1) + 2 * half;
            float u0 = valid ? hs * W_hh[row * HIDN + k0]     : 0.0f;
            float u1 = valid ? hs * W_hh[row * HIDN + k0 + 1] : 0.0f;
            bw[G][s] = v2f{u0, u1};
        }
        // s=3: half=0 -> {h8,h9} weights; half=1 -> {bias, 0} (K=14 bias feature)
        float bb = valid ? rs * (b_ih[row] + b_hh[row]) : 0.0f;
        float u8 = valid ? hs * W_hh[row * HIDN + 8] : 0.0f;
        float u9 = valid ? hs * W_hh[row * HIDN + 9] : 0.0f;
        bw[G][3] = half ? v2f{bb, 0.0f} : v2f{u8, u9};
    }
    float wl[4][3][2], blin[4];
#pragma unroll
    for (int n = 0; n < 4; ++n) {
        blin[n] = b_lin[n];
#pragma unroll
        for (int s = 0; s < 3; ++s)
#pragma unroll
            for (int cc = 0; cc < 2; ++cc) {
                int hidx = 4 * s + 2 * half + cc;
                wl[n][s][cc] = (hidx < HIDN) ? 0.5f * W_lin[n * HIDN + hidx] : 0.0f;
            }
    }

    // ---- state ----
    v8f c_o;
#pragma unroll
    for (int k = 0; k < 8; ++k) c_o[k] = 0.0f;
    v2f a_o[4];
    a_o[1] = v2f{0.0f, 0.0f};
    a_o[2] = v2f{0.0f, 0.0f};
    a_o[3] = half ? v2f{1.0f, 0.0f} : v2f{0.0f, 0.0f};  // bias feature active from t=0

    float* buf = &hbuf[wv][0];
    const float* xrow = x + (m0 + j) * TT * INF;
    float*       orow = out_y + (m0 + j) * (long)TPN * PP * OUTF;

    v2f a_x = *(const v2f*)(xrow + 2 * half);          // software-pipelined x load

#pragma unroll 1
    for (int t = 0; t < TPN; ++t) {
        a_o[0] = a_x;

        v8f hD;
        lstm_cell(a_o, bw, c_o, hD);

        // load next step's x now; 4 inner cells (~hundreds of cycles) cover it
        if (t + 1 < TPN) {
            a_x = *(const v2f*)(xrow + (t + 1) * INF + 2 * half);
            __builtin_prefetch(xrow + (t + 2) * INF, 0, 3);   // WGP-scope prefetch
        }

        transpose_h(hD, buf, j, half, a_o);

        float y[4];
        lin_out(a_o, wl, blin, y);
        if (half == 0) {
            float4 yv = {y[0], y[1], y[2], y[3]};
            *(float4*)(orow + (t * PP + 0) * OUTF) = yv;
        }

        // autoregressive rollout on a state copy
        v8f c_i = c_o;
        v2f a_i[4];
        a_i[1] = a_o[1]; a_i[2] = a_o[2]; a_i[3] = a_o[3];
        for (int p = 1; p < PP; ++p) {
            a_i[0] = half ? v2f{y[2], y[3]} : v2f{y[0], y[1]};
            lstm_cell(a_i, bw, c_i, hD);
            transpose_h(hD, buf, j, half, a_i);
            lin_out(a_i, wl, blin, y);
            if (half == 0) {
                float4 yv = {y[0], y[1], y[2], y[3]};
                *(float4*)(orow + (t * PP + p) * OUTF) = yv;
            }
        }
    }

    // ---- final cell state c: D layout -> out_c[b*HID + j] ----
    if (j < HIDN) {
#pragma unroll
        for (int k = 0; k < 8; ++k)
            out_c[(m0 + k + 8 * half) * HIDN + j] = c_o[k];
    }
}

extern "C" void kernel_launch(void* const* d_in, const int* in_sizes, int n_in,
                              void* d_out, int out_size, void* d_ws, size_t ws_size,
                              hipStream_t stream) {
    const float* x     = (const float*)d_in[0];
    const float* W_ih  = (const float*)d_in[1];
    const float* W_hh  = (const float*)d_in[2];
    const float* b_ih  = (const float*)d_in[3];
    const float* b_hh  = (const float*)d_in[4];
    const float* W_lin = (const float*)d_in[5];
    const float* b_lin = (const float*)d_in[6];
    float* out_y = (float*)d_out;
    float* out_c = out_y + (size_t)BATCH * TPN * PP * OUTF;

    dim3 grid(BATCH / 64);   // 4 waves/block, 16 batch rows per wave
    dim3 block(128);
    arlstm_kernel<<<grid, block, 0, stream>>>(x, W_ih, W_hh, b_ih, b_hh,
                                              W_lin, b_lin, out_y, out_c);
}